// TransFusion_31138512896548
// MI455X (gfx1250) — compile-verified
//
#include <hip/hip_runtime.h>
#include <hip/hip_bf16.h>

typedef float v2f __attribute__((ext_vector_type(2)));
typedef float v8f __attribute__((ext_vector_type(8)));

#define EPS 1e-5f

// ---------------------------------------------------------------------------
// Stats scratch layout (floats, after q/k/v in ws):
//  [0..2]   sum(p_r1)      [3..5]   sumsq(p_r1)
//  [8..10]  A_p            [11..13] B_p           (bn(x) = x*A + B, post gamma/beta)
//  [16..47] sum(w0)        [48..79] sumsq(w0)
//  [80..111] A_w1          [112..143] B_w1
//  [144..147] sum(w1)      [148..151] sumsq(w1)
//  [152..155] A_w2         [156..159] B_w2
// ---------------------------------------------------------------------------

// ---- Kernel 1: q/k/v = x @ W^T + b via V_WMMA_F32_16X16X4_F32 -------------
// One wave handles 16 points. A = x tile (16x4 chunks), B = W^T chunk (4x16).
// f32 A 16x4 layout: lanes 0-15 -> M=lane, VGPR0=K0,VGPR1=K1; lanes 16-31 -> K2,K3.
__global__ void qkv_wmma_kernel(const float* __restrict__ x,
                                const float* __restrict__ Wq, const float* __restrict__ bq,
                                const float* __restrict__ Wk, const float* __restrict__ bk,
                                const float* __restrict__ Wv, const float* __restrict__ bv,
                                float* __restrict__ q, float* __restrict__ k,
                                float* __restrict__ v, int n) {
    int wave = (blockIdx.x * blockDim.x + threadIdx.x) >> 5;
    int lane = threadIdx.x & 31;
    int r0 = wave * 16;
    if (r0 >= n) return;                       // wave-uniform: EXEC stays all-ones
    int m    = lane & 15;
    int koff = (lane >> 4) << 1;               // 0 for lanes 0-15, 2 for 16-31
    int rowadd = (lane >> 4) << 3;             // D rows +8 for upper half-wave

    const float* xrow = x + (size_t)(r0 + m) * 32 + koff;
    v2f a[8];
#pragma unroll
    for (int kc = 0; kc < 8; ++kc)
        a[kc] = *(const v2f*)(xrow + kc * 4);

    const float* Ws[3]   = {Wq, Wk, Wv};
    const float* bs[3]   = {bq, bk, bv};
    float*       outs[3] = {q, k, v};

#pragma unroll
    for (int mat = 0; mat < 3; ++mat) {
#pragma unroll
        for (int t = 0; t < 2; ++t) {
            int ng = t * 16 + m;               // output channel for this lane
            const float* wrow = Ws[mat] + (size_t)ng * 32 + koff;
            v8f acc = {};
#pragma unroll
            for (int kc = 0; kc < 8; ++kc) {
                v2f b = *(const v2f*)(wrow + kc * 4);
                acc = __builtin_amdgcn_wmma_f32_16x16x4_f32(
                        false, a[kc], false, b, (short)0, acc, false, false);
            }
            float bias = bs[mat][ng];
            float* o = outs[mat];
#pragma unroll
            for (int r = 0; r < 8; ++r)
                o[(size_t)(r0 + rowadd + r) * 32 + ng] = acc[r] + bias;
        }
    }
}

// ---- Kernel 2: stats of p_r1 = (p[idx]-p) @ Wp1^T + bp1 --------------------
__global__ void stats_p_kernel(const float* __restrict__ p, const int* __restrict__ idx,
                               const float* __restrict__ Wp1, const float* __restrict__ bp1,
                               float* __restrict__ stats, int n) {
    __shared__ float ls[6];
    if (threadIdx.x < 6) ls[threadIdx.x] = 0.f;
    __syncthreads();
    float W[9], bb[3];
#pragma unroll
    for (int i = 0; i < 9; ++i) W[i] = Wp1[i];
#pragma unroll
    for (int i = 0; i < 3; ++i) bb[i] = bp1[i];
    float s[3] = {0, 0, 0}, ss[3] = {0, 0, 0};
    for (int i = blockIdx.x * blockDim.x + threadIdx.x; i < n;
         i += gridDim.x * blockDim.x) {
        float px = p[i * 3], py = p[i * 3 + 1], pz = p[i * 3 + 2];
        for (int j = 0; j < 8; ++j) {
            int nb = idx[i * 8 + j];
            float dx = p[nb * 3] - px, dy = p[nb * 3 + 1] - py, dz = p[nb * 3 + 2] - pz;
#pragma unroll
            for (int d = 0; d < 3; ++d) {
                float r = fmaf(W[d * 3], dx, fmaf(W[d * 3 + 1], dy, fmaf(W[d * 3 + 2], dz, bb[d])));
                s[d] += r; ss[d] += r * r;
            }
        }
    }
#pragma unroll
    for (int d = 0; d < 3; ++d) { atomicAdd(&ls[d], s[d]); atomicAdd(&ls[3 + d], ss[d]); }
    __syncthreads();
    if (threadIdx.x < 6) atomicAdd(&stats[threadIdx.x], ls[threadIdx.x]);
}

// ---- Finalize BN: sums -> affine (A,B) -------------------------------------
__global__ void finalize_bn_kernel(const float* __restrict__ sums, const float* __restrict__ ssums,
                                   const float* __restrict__ g, const float* __restrict__ b,
                                   float* __restrict__ A, float* __restrict__ B,
                                   int nch, float inv_cnt) {
    int c = threadIdx.x;
    if (c < nch) {
        float m   = sums[c] * inv_cnt;
        float var = fmaf(-m, m, ssums[c] * inv_cnt);
        float a   = g[c] * rsqrtf(var + EPS);
        A[c] = a;
        B[c] = fmaf(-m, a, b[c]);
    }
}

// ---- Kernel 3: stats of w0 = k[idx] - q + p_r2 (warp-per-point, lane=channel)
__global__ void stats_w1_kernel(const float* __restrict__ p, const int* __restrict__ idx,
                                const float* __restrict__ Wp1, const float* __restrict__ bp1,
                                const float* __restrict__ Wp2, const float* __restrict__ bp2,
                                const float* __restrict__ q, const float* __restrict__ k,
                                float* __restrict__ stats, int n) {
    __shared__ float ls[64];
    if (threadIdx.x < 64) ls[threadIdx.x] = 0.f;
    __syncthreads();
    int lane  = threadIdx.x & 31;
    int warp  = (blockIdx.x * blockDim.x + threadIdx.x) >> 5;
    int nwarp = (gridDim.x * blockDim.x) >> 5;
    float w20 = Wp2[lane * 3], w21 = Wp2[lane * 3 + 1], w22 = Wp2[lane * 3 + 2];
    float b2c = bp2[lane];
    float Ap0 = stats[8],  Ap1 = stats[9],  Ap2 = stats[10];
    float Bp0 = stats[11], Bp1 = stats[12], Bp2 = stats[13];
    float W[9];
#pragma unroll
    for (int i = 0; i < 9; ++i) W[i] = Wp1[i];
    float bb0 = bp1[0], bb1 = bp1[1], bb2 = bp1[2];
    float sc = 0.f, ssc = 0.f;
    for (int i = warp; i < n; i += nwarp) {
        float px = p[i * 3], py = p[i * 3 + 1], pz = p[i * 3 + 2];
        float qv = q[(size_t)i * 32 + lane];
        for (int j = 0; j < 8; ++j) {
            int nb = idx[i * 8 + j];
            float dx = p[nb * 3] - px, dy = p[nb * 3 + 1] - py, dz = p[nb * 3 + 2] - pz;
            float r0 = fmaf(W[0], dx, fmaf(W[1], dy, fmaf(W[2], dz, bb0)));
            float r1 = fmaf(W[3], dx, fmaf(W[4], dy, fmaf(W[5], dz, bb1)));
            float r2 = fmaf(W[6], dx, fmaf(W[7], dy, fmaf(W[8], dz, bb2)));
            float t0 = fmaxf(fmaf(r0, Ap0, Bp0), 0.f);
            float t1 = fmaxf(fmaf(r1, Ap1, Bp1), 0.f);
            float t2 = fmaxf(fmaf(r2, Ap2, Bp2), 0.f);
            float pr2 = fmaf(t0, w20, fmaf(t1, w21, fmaf(t2, w22, b2c)));
            float w0 = k[(size_t)nb * 32 + lane] - qv + pr2;
            sc += w0; ssc += w0 * w0;
        }
    }
    atomicAdd(&ls[lane], sc);
    atomicAdd(&ls[32 + lane], ssc);
    __syncthreads();
    if (threadIdx.x < 32)      atomicAdd(&stats[16 + threadIdx.x], ls[threadIdx.x]);
    else if (threadIdx.x < 64) atomicAdd(&stats[48 + threadIdx.x - 32], ls[threadIdx.x]);
}

// ---- Kernel 4: stats of w1 = relu(bn(w0)) @ Ww1^T (thread per (i,j)) -------
__global__ void stats_w2_kernel(const float* __restrict__ p, const int* __restrict__ idx,
                                const float* __restrict__ Wp1, const float* __restrict__ bp1,
                                const float* __restrict__ Wp2, const float* __restrict__ bp2,
                                const float* __restrict__ Ww1, const float* __restrict__ bww1,
                                const float* __restrict__ q, const float* __restrict__ k,
                                float* __restrict__ stats, int n) {
    __shared__ float ls[8];
    if (threadIdx.x < 8) ls[threadIdx.x] = 0.f;
    __syncthreads();
    const float* Ap = stats + 8;  const float* Bp = stats + 11;
    const float* A1 = stats + 80; const float* B1 = stats + 112;
    float s[4] = {0, 0, 0, 0}, ss[4] = {0, 0, 0, 0};
    int total = n * 8;
    for (int t = blockIdx.x * blockDim.x + threadIdx.x; t < total;
         t += gridDim.x * blockDim.x) {
        int i  = t >> 3;
        int nb = idx[t];
        float px = p[i * 3], py = p[i * 3 + 1], pz = p[i * 3 + 2];
        float dx = p[nb * 3] - px, dy = p[nb * 3 + 1] - py, dz = p[nb * 3 + 2] - pz;
        float pr[3];
#pragma unroll
        for (int d = 0; d < 3; ++d) {
            float r = fmaf(Wp1[d * 3], dx, fmaf(Wp1[d * 3 + 1], dy, fmaf(Wp1[d * 3 + 2], dz, bp1[d])));
            pr[d] = fmaxf(fmaf(r, Ap[d], Bp[d]), 0.f);
        }
        const float* kr = k + (size_t)nb * 32;
        const float* qr = q + (size_t)i * 32;
        float w1[4] = {bww1[0], bww1[1], bww1[2], bww1[3]};
#pragma unroll
        for (int c = 0; c < 32; ++c) {
            float pr2 = fmaf(pr[0], Wp2[c * 3], fmaf(pr[1], Wp2[c * 3 + 1], fmaf(pr[2], Wp2[c * 3 + 2], bp2[c])));
            float w0 = kr[c] - qr[c] + pr2;
            float a  = fmaxf(fmaf(w0, A1[c], B1[c]), 0.f);
#pragma unroll
            for (int m2 = 0; m2 < 4; ++m2) w1[m2] = fmaf(a, Ww1[m2 * 32 + c], w1[m2]);
        }
#pragma unroll
        for (int m2 = 0; m2 < 4; ++m2) { s[m2] += w1[m2]; ss[m2] += w1[m2] * w1[m2]; }
    }
#pragma unroll
    for (int m2 = 0; m2 < 4; ++m2) { atomicAdd(&ls[m2], s[m2]); atomicAdd(&ls[4 + m2], ss[m2]); }
    __syncthreads();
    if (threadIdx.x < 4)      atomicAdd(&stats[144 + threadIdx.x], ls[threadIdx.x]);
    else if (threadIdx.x < 8) atomicAdd(&stats[148 + threadIdx.x - 4], ls[threadIdx.x]);
}

// ---- Kernel 5: w2, softmax over neighbors, grouped weighted sum ------------
__global__ void final_kernel(const float* __restrict__ p, const int* __restrict__ idx,
                             const float* __restrict__ Wp1, const float* __restrict__ bp1,
                             const float* __restrict__ Wp2, const float* __restrict__ bp2,
                             const float* __restrict__ Ww1, const float* __restrict__ bww1,
                             const float* __restrict__ Ww2, const float* __restrict__ bww2,
                             const float* __restrict__ q, const float* __restrict__ k,
                             const float* __restrict__ v, const float* __restrict__ stats,
                             float* __restrict__ out, int n) {
    const float* Ap = stats + 8;   const float* Bp = stats + 11;
    const float* A1 = stats + 80;  const float* B1 = stats + 112;
    const float* A2 = stats + 152; const float* B2 = stats + 156;
    for (int i = blockIdx.x * blockDim.x + threadIdx.x; i < n;
         i += gridDim.x * blockDim.x) {
        float px = p[i * 3], py = p[i * 3 + 1], pz = p[i * 3 + 2];
        const float* qr = q + (size_t)i * 32;
        int   nbj[8];
        float prj[8][3];
        float w2[8][4];
#pragma unroll
        for (int j = 0; j < 8; ++j) {
            int nb = idx[i * 8 + j];
            nbj[j] = nb;
            float dx = p[nb * 3] - px, dy = p[nb * 3 + 1] - py, dz = p[nb * 3 + 2] - pz;
#pragma unroll
            for (int d = 0; d < 3; ++d) {
                float r = fmaf(Wp1[d * 3], dx, fmaf(Wp1[d * 3 + 1], dy, fmaf(Wp1[d * 3 + 2], dz, bp1[d])));
                prj[j][d] = fmaxf(fmaf(r, Ap[d], Bp[d]), 0.f);
            }
            const float* kr = k + (size_t)nb * 32;
            float w1[4] = {bww1[0], bww1[1], bww1[2], bww1[3]};
#pragma unroll
            for (int c = 0; c < 32; ++c) {
                float pr2 = fmaf(prj[j][0], Wp2[c * 3],
                            fmaf(prj[j][1], Wp2[c * 3 + 1],
                            fmaf(prj[j][2], Wp2[c * 3 + 2], bp2[c])));
                float w0 = kr[c] - qr[c] + pr2;
                float a  = fmaxf(fmaf(w0, A1[c], B1[c]), 0.f);
#pragma unroll
                for (int m2 = 0; m2 < 4; ++m2) w1[m2] = fmaf(a, Ww1[m2 * 32 + c], w1[m2]);
            }
            float a0 = fmaxf(fmaf(w1[0], A2[0], B2[0]), 0.f);
            float a1 = fmaxf(fmaf(w1[1], A2[1], B2[1]), 0.f);
            float a2 = fmaxf(fmaf(w1[2], A2[2], B2[2]), 0.f);
            float a3 = fmaxf(fmaf(w1[3], A2[3], B2[3]), 0.f);
#pragma unroll
            for (int m2 = 0; m2 < 4; ++m2)
                w2[j][m2] = fmaf(a0, Ww2[m2 * 4],
                            fmaf(a1, Ww2[m2 * 4 + 1],
                            fmaf(a2, Ww2[m2 * 4 + 2],
                            fmaf(a3, Ww2[m2 * 4 + 3], bww2[m2]))));
        }
        // softmax over the 8 neighbors, per output sub-channel
#pragma unroll
        for (int m2 = 0; m2 < 4; ++m2) {
            float mx = w2[0][m2];
#pragma unroll
            for (int j = 1; j < 8; ++j) mx = fmaxf(mx, w2[j][m2]);
            float sum = 0.f;
#pragma unroll
            for (int j = 0; j < 8; ++j) { float e = __expf(w2[j][m2] - mx); w2[j][m2] = e; sum += e; }
            float inv = 1.f / sum;
#pragma unroll
            for (int j = 0; j < 8; ++j) w2[j][m2] *= inv;
        }
        float acc[32];
#pragma unroll
        for (int c = 0; c < 32; ++c) acc[c] = 0.f;
#pragma unroll
        for (int j = 0; j < 8; ++j) {
            const float* vr = v + (size_t)nbj[j] * 32;
#pragma unroll
            for (int c = 0; c < 32; ++c) {
                float pr2 = fmaf(prj[j][0], Wp2[c * 3],
                            fmaf(prj[j][1], Wp2[c * 3 + 1],
                            fmaf(prj[j][2], Wp2[c * 3 + 2], bp2[c])));
                acc[c] = fmaf(vr[c] + pr2, w2[j][c & 3], acc[c]);
            }
        }
#pragma unroll
        for (int c = 0; c < 32; ++c) out[(size_t)i * 32 + c] = acc[c];
    }
}

// ---------------------------------------------------------------------------
extern "C" void kernel_launch(void* const* d_in, const int* in_sizes, int n_in,
                              void* d_out, int out_size, void* d_ws, size_t ws_size,
                              hipStream_t stream) {
    const float* p    = (const float*)d_in[0];
    const float* x    = (const float*)d_in[1];
    const int*   idx  = (const int*)  d_in[2];
    const float* Wq   = (const float*)d_in[3];
    const float* bq   = (const float*)d_in[4];
    const float* Wk   = (const float*)d_in[5];
    const float* bk   = (const float*)d_in[6];
    const float* Wv   = (const float*)d_in[7];
    const float* bv   = (const float*)d_in[8];
    const float* Wp1  = (const float*)d_in[9];
    const float* bp1  = (const float*)d_in[10];
    const float* gp   = (const float*)d_in[11];
    const float* bp_  = (const float*)d_in[12];
    const float* Wp2  = (const float*)d_in[13];
    const float* bp2  = (const float*)d_in[14];
    const float* gw1  = (const float*)d_in[15];
    const float* bw1  = (const float*)d_in[16];
    const float* Ww1  = (const float*)d_in[17];
    const float* bww1 = (const float*)d_in[18];
    const float* gw2  = (const float*)d_in[19];
    const float* bw2  = (const float*)d_in[20];
    const float* Ww2  = (const float*)d_in[21];
    const float* bww2 = (const float*)d_in[22];

    int n = in_sizes[0] / 3;
    float inv_cnt = 1.0f / (float)((long long)n * 8);

    float* q     = (float*)d_ws;
    float* k     = q + (size_t)n * 32;
    float* v     = k + (size_t)n * 32;
    float* stats = v + (size_t)n * 32;

    hipMemsetAsync(stats, 0, 160 * sizeof(float), stream);

    int qkvBlocks = (n / 16 + 7) / 8;   // 8 waves of 32 threads per 256-thread block
    qkv_wmma_kernel<<<qkvBlocks, 256, 0, stream>>>(x, Wq, bq, Wk, bk, Wv, bv, q, k, v, n);

    stats_p_kernel<<<1024, 256, 0, stream>>>(p, idx, Wp1, bp1, stats, n);
    finalize_bn_kernel<<<1, 32, 0, stream>>>(stats + 0, stats + 3, gp, bp_,
                                             stats + 8, stats + 11, 3, inv_cnt);

    stats_w1_kernel<<<1024, 256, 0, stream>>>(p, idx, Wp1, bp1, Wp2, bp2, q, k, stats, n);
    finalize_bn_kernel<<<1, 32, 0, stream>>>(stats + 16, stats + 48, gw1, bw1,
                                             stats + 80, stats + 112, 32, inv_cnt);

    stats_w2_kernel<<<4096, 256, 0, stream>>>(p, idx, Wp1, bp1, Wp2, bp2, Ww1, bww1,
                                              q, k, stats, n);
    finalize_bn_kernel<<<1, 32, 0, stream>>>(stats + 144, stats + 148, gw2, bw2,
                                             stats + 152, stats + 156, 4, inv_cnt);

    final_kernel<<<2048, 256, 0, stream>>>(p, idx, Wp1, bp1, Wp2, bp2, Ww1, bww1,
                                           Ww2, bww2, q, k, v, stats,
                                           (float*)d_out, n);
}